// FuzzyGraphConv_31318901522778
// MI455X (gfx1250) — compile-verified
//
#include <hip/hip_runtime.h>

#define N_NODES 50000
#define N_EDGES 800000
#define D_IN    512
#define D_OUT   256
#define EPW     64   // edges per wave in spmm

typedef float v2f __attribute__((ext_vector_type(2)));
typedef float v8f __attribute__((ext_vector_type(8)));

#if defined(__gfx1250__) && __has_builtin(__builtin_amdgcn_global_load_async_to_lds_b128)
#define HAVE_ASYNC_LDS 1
typedef int v4i_vec __attribute__((vector_size(16)));
typedef __attribute__((address_space(1))) v4i_vec* as1_v4ip;
typedef __attribute__((address_space(3))) v4i_vec* as3_v4ip;
#else
#define HAVE_ASYNC_LDS 0
#endif

// ---------------------------------------------------------------------------
// Zero the center accumulator (d_ws) — harness poisons ws with 0xAA.
// ---------------------------------------------------------------------------
__global__ __launch_bounds__(256) void zero_f4(float4* __restrict__ p, int n4) {
    int i = blockIdx.x * blockDim.x + threadIdx.x;
    if (i < n4) {
        float4 z; z.x = 0.f; z.y = 0.f; z.z = 0.f; z.w = 0.f;
        p[i] = z;
    }
}

// ---------------------------------------------------------------------------
// Fused triple GEMM via V_WMMA_F32_16X16X4_F32 with LDS-staged weights.
//   xb   =  x  @ w_b            (staged into out1 slot of d_out)
//   outL = -(|x| @ relu(w_a))   (partial of out2, center added later)
//   outR =  (|x| @ relu(w_c))   (partial of out3)
//
// Block = 8 waves sharing one N-tile. The 512x16 slice of all three weight
// matrices (96 KB) is staged into LDS once per block — via
// GLOBAL_LOAD_ASYNC_TO_LDS_B128 (ASYNCcnt) when available — then relu'd in
// place for w_a/w_c. Each wave computes TWO 16x16 M-tiles so every set of
// 6 ds_loads feeds 6 WMMAs. A-rows are clamped and stores guarded for the
// M tail, keeping EXEC all-1s around every WMMA.
// ---------------------------------------------------------------------------
__global__ __launch_bounds__(256) void gemm3_wmma(
    const float* __restrict__ x,
    const float* __restrict__ wb,
    const float* __restrict__ wa,
    const float* __restrict__ wc,
    float* __restrict__ xb,
    float* __restrict__ outL,
    float* __restrict__ outR)
{
    __shared__ float wsm[3 * D_IN * 16];            // 96 KB: [mat][k][16]

    const int tid     = threadIdx.x;
    const int tileN   = blockIdx.x & 15;            // D_OUT/16 = 16 N-tiles
    const int mGroup  = blockIdx.x >> 4;            // 16 M-tiles per block
    const int colBase = tileN * 16;

    // ---- stage the three 512x16 weight slices into LDS ----
    // 2048 float4 per matrix; 256 threads -> 8 float4/thread/matrix.
    const float* const srcs[3] = { wb, wa, wc };
    #pragma unroll
    for (int mat = 0; mat < 3; ++mat) {
        const float* src = srcs[mat];
        #pragma unroll
        for (int t = 0; t < 8; ++t) {
            const int q  = tid + t * 256;           // quad index 0..2047
            const int k  = q >> 2;                  // weight row
            const int qi = q & 3;                   // quad within the 16 cols
            const size_t gOff = (size_t)k * D_OUT + colBase + qi * 4;
            const int    lOff = mat * (D_IN * 16) + q * 4;
#if HAVE_ASYNC_LDS
            __builtin_amdgcn_global_load_async_to_lds_b128(
                (as1_v4ip)(src + gOff), (as3_v4ip)(wsm + lOff), 0, 0);
#else
            *(float4*)(wsm + lOff) = *(const float4*)(src + gOff);
#endif
        }
    }
#if HAVE_ASYNC_LDS
    asm volatile("s_wait_asynccnt 0x0" ::: "memory");
#endif
    __syncthreads();

    // ---- relu w_a / w_c slices in place (4096 float4 / 256 threads = 16) ----
    {
        float4* w4 = (float4*)(wsm + D_IN * 16);    // start of wa region
        #pragma unroll
        for (int t = 0; t < 16; ++t) {
            const int q = tid + t * 256;
            float4 v = w4[q];
            v.x = __builtin_fmaxf(v.x, 0.f); v.y = __builtin_fmaxf(v.y, 0.f);
            v.z = __builtin_fmaxf(v.z, 0.f); v.w = __builtin_fmaxf(v.w, 0.f);
            w4[q] = v;
        }
    }
    __syncthreads();

    // ---- WMMA main loop: 2 M-tiles per wave, 3 outputs -> 6 accumulators ----
    const int lane = tid & 31;
    const int wv   = tid >> 5;
    const int half = lane >> 4;                     // 0 or 1
    const int l16  = lane & 15;

    const int tm0 = mGroup * 16 + wv * 2;
    const int tm1 = tm0 + 1;
    // Clamp A-row loads for the M tail (3136*16 padded vs 3125 real tiles).
    const int r0 = min(tm0 * 16 + l16, N_NODES - 1);
    const int r1 = min(tm1 * 16 + l16, N_NODES - 1);

    // A layout (32-bit 16x4): lanes 0-15 hold K=k+0,k+1; lanes 16-31 K=k+2,k+3
    const float* x0 = x + (size_t)r0 * D_IN + half * 2;
    const float* x1 = x + (size_t)r1 * D_IN + half * 2;
    // B layout (32-bit 4x16): .x = row k+half*2, .y = next row; N = lane&15
    const float* Bb = &wsm[0 * (D_IN * 16) + half * 32 + l16];
    const float* Ba = &wsm[1 * (D_IN * 16) + half * 32 + l16];
    const float* Bc = &wsm[2 * (D_IN * 16) + half * 32 + l16];

    v8f accB0 = {0.f,0.f,0.f,0.f,0.f,0.f,0.f,0.f};
    v8f accL0 = accB0, accR0 = accB0;
    v8f accB1 = accB0, accL1 = accB0, accR1 = accB0;

    #pragma unroll 2
    for (int k = 0; k < D_IN; k += 4) {
        v2f a0; a0.x = x0[k]; a0.y = x0[k + 1];
        v2f a1; a1.x = x1[k]; a1.y = x1[k + 1];
        v2f u0; u0.x = __builtin_fabsf(a0.x); u0.y = __builtin_fabsf(a0.y);
        v2f u1; u1.x = __builtin_fabsf(a1.x); u1.y = __builtin_fabsf(a1.y);

        const int ko = k * 16;
        v2f bb; bb.x = Bb[ko]; bb.y = Bb[ko + 16];
        v2f ba; ba.x = Ba[ko]; ba.y = Ba[ko + 16];
        v2f bc; bc.x = Bc[ko]; bc.y = Bc[ko + 16];

        accB0 = __builtin_amdgcn_wmma_f32_16x16x4_f32(false, a0, false, bb, (short)0, accB0, false, false);
        accB1 = __builtin_amdgcn_wmma_f32_16x16x4_f32(false, a1, false, bb, (short)0, accB1, false, false);
        accL0 = __builtin_amdgcn_wmma_f32_16x16x4_f32(false, u0, false, ba, (short)0, accL0, false, false);
        accL1 = __builtin_amdgcn_wmma_f32_16x16x4_f32(false, u1, false, ba, (short)0, accL1, false, false);
        accR0 = __builtin_amdgcn_wmma_f32_16x16x4_f32(false, u0, false, bc, (short)0, accR0, false, false);
        accR1 = __builtin_amdgcn_wmma_f32_16x16x4_f32(false, u1, false, bc, (short)0, accR1, false, false);
    }

    // C/D layout (32-bit 16x16): VGPR v -> M = v + 8*half, N = lane&15.
    const int colB = colBase + l16;
    #pragma unroll
    for (int v = 0; v < 8; ++v) {
        const int m0 = tm0 * 16 + v + half * 8;
        if (m0 < N_NODES) {
            const size_t o = (size_t)m0 * D_OUT + colB;
            xb[o]   =  accB0[v];
            outL[o] = -accL0[v];
            outR[o] =  accR0[v];
        }
        const int m1 = tm1 * 16 + v + half * 8;
        if (m1 < N_NODES) {
            const size_t o = (size_t)m1 * D_OUT + colB;
            xb[o]   =  accB1[v];
            outL[o] = -accL1[v];
            outR[o] =  accR1[v];
        }
    }
}

// ---------------------------------------------------------------------------
// SpMM: center[r,:] += vals[e] * xb[cols[e],:] for sorted rows.
// One wave owns EPW contiguous edges; 32 lanes x 8 floats cover the 256-wide
// row; accumulate in registers while the row id is unchanged, flush with
// global_atomic_add_f32 on row boundaries. xb (51 MB) is L2-resident.
// ---------------------------------------------------------------------------
__global__ __launch_bounds__(256) void spmm_acc(
    const int*   __restrict__ rows,
    const int*   __restrict__ cols,
    const float* __restrict__ vals,
    const float* __restrict__ xb,
    float*       __restrict__ center)
{
    const int lane = threadIdx.x & 31;
    const int wave = (blockIdx.x * blockDim.x + threadIdx.x) >> 5;
    long e0 = (long)wave * EPW;
    if (e0 >= N_EDGES) return;
    long e1 = e0 + EPW;
    if (e1 > N_EDGES) e1 = N_EDGES;

    float acc[8] = {0.f,0.f,0.f,0.f,0.f,0.f,0.f,0.f};
    int cur = rows[e0];

    for (long e = e0; e < e1; ++e) {
        const int r = rows[e];           // uniform across the wave
        if (r != cur) {
            float* dst = center + (size_t)cur * D_OUT + lane * 8;
            #pragma unroll
            for (int j = 0; j < 8; ++j) {
                atomicAdd(dst + j, acc[j]);
                acc[j] = 0.f;
            }
            cur = r;
        }
        const int   c = cols[e];
        const float v = vals[e];
        if (e + 1 < e1)                  // emits global_prefetch_b8
            __builtin_prefetch(xb + (size_t)cols[e + 1] * D_OUT + lane * 8, 0, 3);
        const float4* src = (const float4*)(xb + (size_t)c * D_OUT) + lane * 2;
        const float4 p0 = src[0];
        const float4 p1 = src[1];
        acc[0] += v * p0.x; acc[1] += v * p0.y; acc[2] += v * p0.z; acc[3] += v * p0.w;
        acc[4] += v * p1.x; acc[5] += v * p1.y; acc[6] += v * p1.z; acc[7] += v * p1.w;
    }
    float* dst = center + (size_t)cur * D_OUT + lane * 8;
    #pragma unroll
    for (int j = 0; j < 8; ++j) atomicAdd(dst + j, acc[j]);
}

// ---------------------------------------------------------------------------
// Epilogue: c = center + b_b
//   out1 = c                      (overwrites the xb staging area)
//   out2 = out2 + c - relu(b_a)   (out2 holds -|x|@relu(w_a))
//   out3 = out3 + c + relu(b_c)   (out3 holds  |x|@relu(w_c))
// ---------------------------------------------------------------------------
__global__ __launch_bounds__(256) void finalize(
    const float* __restrict__ center,
    const float* __restrict__ b_b,
    const float* __restrict__ b_a,
    const float* __restrict__ b_c,
    float* __restrict__ out)
{
    const size_t total4 = (size_t)N_NODES * (D_OUT / 4);
    const size_t i = (size_t)blockIdx.x * blockDim.x + threadIdx.x;
    if (i >= total4) return;
    const int cq = (int)(i & (D_OUT / 4 - 1));   // column quad within a row

    float4 c4 = ((const float4*)center)[i];
    const float4 bb = ((const float4*)b_b)[cq];
    float4 ba = ((const float4*)b_a)[cq];
    float4 bc = ((const float4*)b_c)[cq];
    ba.x = __builtin_fmaxf(ba.x, 0.f); ba.y = __builtin_fmaxf(ba.y, 0.f);
    ba.z = __builtin_fmaxf(ba.z, 0.f); ba.w = __builtin_fmaxf(ba.w, 0.f);
    bc.x = __builtin_fmaxf(bc.x, 0.f); bc.y = __builtin_fmaxf(bc.y, 0.f);
    bc.z = __builtin_fmaxf(bc.z, 0.f); bc.w = __builtin_fmaxf(bc.w, 0.f);

    c4.x += bb.x; c4.y += bb.y; c4.z += bb.z; c4.w += bb.w;

    float4* o1 = (float4*)out;
    float4* o2 = o1 + total4;
    float4* o3 = o2 + total4;

    float4 l = o2[i];
    float4 r = o3[i];
    l.x += c4.x - ba.x; l.y += c4.y - ba.y; l.z += c4.z - ba.z; l.w += c4.w - ba.w;
    r.x += c4.x + bc.x; r.y += c4.y + bc.y; r.z += c4.z + bc.z; r.w += c4.w + bc.w;

    o1[i] = c4;
    o2[i] = l;
    o3[i] = r;
}

// ---------------------------------------------------------------------------
extern "C" void kernel_launch(void* const* d_in, const int* in_sizes, int n_in,
                              void* d_out, int out_size, void* d_ws, size_t ws_size,
                              hipStream_t stream) {
    const float* x    = (const float*)d_in[0];
    const int*   rows = (const int*)  d_in[1];
    const int*   cols = (const int*)  d_in[2];
    const float* vals = (const float*)d_in[3];
    const float* wb   = (const float*)d_in[4];
    const float* wa   = (const float*)d_in[5];
    const float* wc   = (const float*)d_in[6];
    const float* bb   = (const float*)d_in[7];
    const float* ba   = (const float*)d_in[8];
    const float* bc   = (const float*)d_in[9];

    float* out  = (float*)d_out;
    const size_t mat = (size_t)N_NODES * D_OUT;      // 12.8M floats
    float* xb   = out;                 // stage x@w_b in the out1 slot
    float* outL = out + mat;           // out2 slot: -(|x|@relu(w_a)) partial
    float* outR = out + 2 * mat;       // out3 slot:  (|x|@relu(w_c)) partial
    float* center = (float*)d_ws;      // 51.2 MB atomic accumulator

    // 1) zero the center accumulator
    const int n4 = (int)(mat / 4);
    zero_f4<<<(n4 + 255) / 256, 256, 0, stream>>>((float4*)center, n4);

    // 2) fused triple GEMM: 16 M-tiles x 1 N-tile per block
    const int mGroups = (N_NODES / 16 + 15) / 16;    // ceil(3125/16) = 196
    gemm3_wmma<<<mGroups * (D_OUT / 16), 256, 0, stream>>>(
        x, wb, wa, wc, xb, outL, outR);

    // 3) SpMM segment-sum into center
    const int nwaves = (N_EDGES + EPW - 1) / EPW;    // 12500
    spmm_acc<<<(nwaves * 32 + 255) / 256, 256, 0, stream>>>(
        rows, cols, vals, xb, center);

    // 4) biases + combine, write all three outputs
    const size_t total4 = mat / 4;
    finalize<<<(int)((total4 + 255) / 256), 256, 0, stream>>>(
        center, bb, ba, bc, out);
}